// GATLayer_34325378629823
// MI455X (gfx1250) — compile-verified
//
#include <hip/hip_runtime.h>
#include <math.h>

typedef __attribute__((ext_vector_type(2))) float v2f;
typedef __attribute__((ext_vector_type(8))) float v8f;

#define IN_DIM   128
#define OUT_DIM  64
#define EDGE_DIM 8

// ---------------------------------------------------------------------------
// exact float atomic-max via sign-split int/uint atomics (works with -inf init)
// ---------------------------------------------------------------------------
__device__ __forceinline__ void atomicMaxF(float* addr, float val) {
    if (!__builtin_signbitf(val)) {
        atomicMax((int*)addr, __float_as_int(val));
    } else {
        atomicMin((unsigned int*)addr, __float_as_uint(val));
    }
}

// ---------------------------------------------------------------------------
// K0: init accumulators (d_out / d_ws are poisoned before timing)
// ---------------------------------------------------------------------------
__global__ void k_init(float* __restrict__ out, float* __restrict__ norm,
                       float* __restrict__ max_e, int n_nodes) {
    int i = blockIdx.x * blockDim.x + threadIdx.x;
    if (i < n_nodes * OUT_DIM) out[i] = 0.0f;
    if (i < n_nodes) {
        norm[i]  = 0.0f;
        max_e[i] = -__builtin_inff();
    }
}

// ---------------------------------------------------------------------------
// K1: wh = h @ W + b   via V_WMMA_F32_16X16X4_F32
// one wave computes a 16x64 strip: 4 accumulators, K-loop of 32 x (16x16x4)
// A 16x4 layout : lanes 0-15 -> M=lane, vgpr{0,1}=K{0,1}; lanes 16-31 -> K{2,3}
// B 4x16 layout : lanes 0-15 -> N=lane, vgpr{0,1}=K{0,1}; lanes 16-31 -> K{2,3}
// C/D 16x16     : vgpr v -> M=v (lanes 0-15) / M=v+8 (lanes 16-31), N=lane%16
// ---------------------------------------------------------------------------
__global__ void __launch_bounds__(256)
k_wmma_transform(const float* __restrict__ h, const float* __restrict__ w_w,
                 const float* __restrict__ w_b, float* __restrict__ wh,
                 int n_nodes) {
    const int wave = threadIdx.x >> 5;
    const int lane = threadIdx.x & 31;
    const int tile = blockIdx.x * 8 + wave;        // 16-row tile of h
    const int m0   = tile * 16;
    if (m0 >= n_nodes) return;

    const int lh = lane & 15;
    const int hi = lane >> 4;

    int arow = m0 + lh;
    if (arow >= n_nodes) arow = n_nodes - 1;       // clamp (harmless pad rows)
    const float* __restrict__ aptr = h + (size_t)arow * IN_DIM + 2 * hi;

    v8f acc0 = {}, acc1 = {}, acc2 = {}, acc3 = {};

    for (int k0 = 0; k0 < IN_DIM; k0 += 4) {
        v2f a;
        a.x = aptr[k0];
        a.y = aptr[k0 + 1];

        const float* __restrict__ bptr = w_w + (size_t)(k0 + 2 * hi) * OUT_DIM + lh;
        v2f b0, b1, b2, b3;
        b0.x = bptr[0];   b0.y = bptr[OUT_DIM + 0];
        b1.x = bptr[16];  b1.y = bptr[OUT_DIM + 16];
        b2.x = bptr[32];  b2.y = bptr[OUT_DIM + 32];
        b3.x = bptr[48];  b3.y = bptr[OUT_DIM + 48];

        acc0 = __builtin_amdgcn_wmma_f32_16x16x4_f32(false, a, false, b0,
                                                     (short)0, acc0, false, false);
        acc1 = __builtin_amdgcn_wmma_f32_16x16x4_f32(false, a, false, b1,
                                                     (short)0, acc1, false, false);
        acc2 = __builtin_amdgcn_wmma_f32_16x16x4_f32(false, a, false, b2,
                                                     (short)0, acc2, false, false);
        acc3 = __builtin_amdgcn_wmma_f32_16x16x4_f32(false, a, false, b3,
                                                     (short)0, acc3, false, false);
    }

    const float bias0 = w_b[lh];
    const float bias1 = w_b[16 + lh];
    const float bias2 = w_b[32 + lh];
    const float bias3 = w_b[48 + lh];

    const int mbase = m0 + 8 * hi;
#pragma unroll
    for (int v = 0; v < 8; ++v) {
        const int m = mbase + v;
        if (m < n_nodes) {
            float* __restrict__ o = wh + (size_t)m * OUT_DIM + lh;
            o[0]  = acc0[v] + bias0;
            o[16] = acc1[v] + bias1;
            o[32] = acc2[v] + bias2;
            o[48] = acc3[v] + bias3;
        }
    }
}

// ---------------------------------------------------------------------------
// K2: per-node attention scores s1 = wh . a1 , s2 = wh . a2
// ---------------------------------------------------------------------------
__global__ void k_scores(const float* __restrict__ wh, const float* __restrict__ a_w,
                         float* __restrict__ s1, float* __restrict__ s2,
                         int n_nodes) {
    int i = blockIdx.x * blockDim.x + threadIdx.x;
    if (i >= n_nodes) return;
    const float4* __restrict__ row = (const float4*)(wh + (size_t)i * OUT_DIM);
    const float4* __restrict__ p1  = (const float4*)(a_w);
    const float4* __restrict__ p2  = (const float4*)(a_w + OUT_DIM);
    float acc1 = 0.0f, acc2 = 0.0f;
#pragma unroll
    for (int j = 0; j < OUT_DIM / 4; ++j) {
        float4 w = row[j];
        float4 p = p1[j];
        float4 q = p2[j];
        acc1 += w.x * p.x + w.y * p.y + w.z * p.z + w.w * p.w;
        acc2 += w.x * q.x + w.y * q.y + w.z * q.z + w.w * q.w;
    }
    s1[i] = acc1;
    s2[i] = acc2;
}

// ---------------------------------------------------------------------------
// K3: per-edge logits + segment-max (atomic)
// ---------------------------------------------------------------------------
__global__ void k_edge_logits(const int* __restrict__ ei, const float* __restrict__ ea,
                              const float* __restrict__ s1, const float* __restrict__ s2,
                              const float* __restrict__ a_w, const float* __restrict__ a_b,
                              float* __restrict__ e, float* __restrict__ max_e,
                              int n_edges) {
    int i = blockIdx.x * blockDim.x + threadIdx.x;
    if (i >= n_edges) return;
    const int r = ei[i];
    const int c = ei[n_edges + i];

    const float4* __restrict__ e4 = (const float4*)(ea + (size_t)i * EDGE_DIM);
    const float4* __restrict__ a3 = (const float4*)(a_w + 2 * OUT_DIM);
    float4 x0 = e4[0], x1 = e4[1];
    float4 w0 = a3[0], w1 = a3[1];
    float dot = x0.x * w0.x + x0.y * w0.y + x0.z * w0.z + x0.w * w0.w
              + x1.x * w1.x + x1.y * w1.y + x1.z * w1.z + x1.w * w1.w;

    float v = s1[r] + s2[c] + dot + a_b[0];
    v = (v > 0.0f) ? v : 0.01f * v;          // leaky_relu, slope 0.01
    e[i] = v;
    atomicMaxF(max_e + r, v);
}

// ---------------------------------------------------------------------------
// K4: softmax numerator scatter: one wave per edge, 2 coalesced atomics/lane
// ---------------------------------------------------------------------------
__global__ void __launch_bounds__(256)
k_edge_scatter(const int* __restrict__ ei, const float* __restrict__ e,
               const float* __restrict__ max_e, const float* __restrict__ wh,
               float* __restrict__ out, float* __restrict__ norm, int n_edges) {
    const int lane = threadIdx.x & 31;
    const int edge = blockIdx.x * 8 + (threadIdx.x >> 5);
    if (edge >= n_edges) return;

    const int r = ei[edge];
    const int c = ei[n_edges + edge];
    const float alpha = expf(e[edge] - max_e[r]);   // same-address loads broadcast

    const float* __restrict__ src = wh  + (size_t)c * OUT_DIM;
    float*       __restrict__ dst = out + (size_t)r * OUT_DIM;
    atomicAdd(dst + lane,      alpha * src[lane]);
    atomicAdd(dst + 32 + lane, alpha * src[32 + lane]);
    if (lane == 0) atomicAdd(norm + r, alpha);
}

// ---------------------------------------------------------------------------
// K5: normalize
// ---------------------------------------------------------------------------
__global__ void k_normalize(float* __restrict__ out, const float* __restrict__ norm,
                            int n_nodes) {
    int i = blockIdx.x * blockDim.x + threadIdx.x;
    if (i >= n_nodes * OUT_DIM) return;
    out[i] = out[i] / (norm[i / OUT_DIM] + 1e-8f);
}

// ---------------------------------------------------------------------------
extern "C" void kernel_launch(void* const* d_in, const int* in_sizes, int n_in,
                              void* d_out, int out_size, void* d_ws, size_t ws_size,
                              hipStream_t stream) {
    const float* h   = (const float*)d_in[0];
    const int*   ei  = (const int*)  d_in[1];
    const float* ea  = (const float*)d_in[2];
    const float* w_w = (const float*)d_in[3];
    const float* w_b = (const float*)d_in[4];
    const float* a_w = (const float*)d_in[5];
    const float* a_b = (const float*)d_in[6];
    float* out = (float*)d_out;

    const int n_nodes = in_sizes[0] / IN_DIM;
    const int n_edges = in_sizes[2] / EDGE_DIM;

    // workspace layout (floats): wh | s1 | s2 | max_e | norm | e  (~33 MB)
    float* wh    = (float*)d_ws;
    float* s1    = wh    + (size_t)n_nodes * OUT_DIM;
    float* s2    = s1    + n_nodes;
    float* max_e = s2    + n_nodes;
    float* norm  = max_e + n_nodes;
    float* e     = norm  + n_nodes;

    const int total = n_nodes * OUT_DIM;

    k_init<<<(total + 255) / 256, 256, 0, stream>>>(out, norm, max_e, n_nodes);

    const int n_tiles = (n_nodes + 15) / 16;
    k_wmma_transform<<<(n_tiles + 7) / 8, 256, 0, stream>>>(h, w_w, w_b, wh, n_nodes);

    k_scores<<<(n_nodes + 255) / 256, 256, 0, stream>>>(wh, a_w, s1, s2, n_nodes);

    k_edge_logits<<<(n_edges + 255) / 256, 256, 0, stream>>>(ei, ea, s1, s2, a_w, a_b,
                                                             e, max_e, n_edges);

    k_edge_scatter<<<(n_edges + 7) / 8, 256, 0, stream>>>(ei, e, max_e, wh,
                                                          out, norm, n_edges);

    k_normalize<<<(total + 255) / 256, 256, 0, stream>>>(out, norm, n_nodes);
}